// MyModel_40218073759792
// MI455X (gfx1250) — compile-verified
//
#include <hip/hip_runtime.h>

typedef __attribute__((ext_vector_type(2))) float v2f;
typedef __attribute__((ext_vector_type(8))) float v8f;

#define IN_DIM 225
#define NS 7
#define HIDN 4
#define NCOL 28
#define KSTEPS 57                      // ceil(225/4)
#define ROWS_PER_BLOCK 128             // 8 waves x 16 rows
#define THREADS 256
#define WFRAG_FLOATS (KSTEPS * 128)    // [k][t][lane][j] = 7296 floats

// ---------------------------------------------------------------------------
// Prep kernel: repack W1 into WMMA B-fragment order and fold W2/b2/scales into
// per-column epilogue constants.
//   wfrag[((k*2 + t)*32 + lane)*2 + j] = Wc[K][N],  K = 4k + 2*(lane>>4) + j,
//                                        N = 16t + (lane&15)
//   Wc[i][n] = W1[s= n/4][i][h = n%4], zero padded for K>=225 or N>=28.
// ---------------------------------------------------------------------------
__global__ __launch_bounds__(THREADS) void prep_kernel(
    const float* __restrict__ W1, const float* __restrict__ b1,
    const float* __restrict__ W2, const float* __restrict__ b2,
    float* __restrict__ ws)
{
  float* wfrag = ws;
  float* scl = ws + WFRAG_FLOATS;
  float* b1c = scl + 32;
  float* cc  = b1c + 32;
  float* bcv = cc + 32;

  const int tid = threadIdx.x;
  for (int e = tid; e < WFRAG_FLOATS; e += THREADS) {
    const int j    = e & 1;
    const int lane = (e >> 1) & 31;
    const int t    = (e >> 6) & 1;
    const int k    = e >> 7;
    const int K = 4 * k + 2 * (lane >> 4) + j;
    const int N = 16 * t + (lane & 15);
    float v = 0.0f;
    if (K < IN_DIM && N < NCOL) {
      const int s = N >> 2, h = N & 3;
      v = W1[(s * IN_DIM + K) * HIDN + h];
    }
    wfrag[e] = v;
  }

  if (tid < 32) {
    const int n = tid;
    float vs = 0.0f, vb = 0.0f, vc = 0.0f;
    if (n < NCOL) {
      const int s = n >> 2, h = n & 3;
      vs = exp2f(-(float)s);            // scale_s = 0.5^s
      vb = b1[s * HIDN + h];
      float w2sum = 0.0f;
      for (int k2 = 0; k2 < HIDN; ++k2) w2sum += W2[(s * HIDN + h) * HIDN + k2];
      vc = w2sum * exp2f((float)s) * (1.0f / (float)(NS * HIDN));
    }
    scl[n] = vs; b1c[n] = vb; cc[n] = vc;
  }

  if (tid == 0) {
    float acc = 0.0f;
    for (int s = 0; s < NS; ++s) {
      const float inv = exp2f((float)s);
      for (int k2 = 0; k2 < HIDN; ++k2) acc += b2[s * HIDN + k2] * inv;
    }
    bcv[0] = acc * (1.0f / (float)(NS * HIDN));
  }
}

// ---------------------------------------------------------------------------
// Main kernel: one wave32 per 16-row tile. V_WMMA_F32_16X16X4_F32, K swept in
// 57 steps, two N-tiles (cols 0..15 / 16..31) accumulated in the same pass so
// x is streamed exactly once (HBM-bound: ~236 MB total).
// ---------------------------------------------------------------------------
__global__ __launch_bounds__(THREADS) void mlp_wmma_kernel(
    const float* __restrict__ x, const float* __restrict__ ws,
    float* __restrict__ out)
{
  const float* __restrict__ wfrag = ws;
  const float* __restrict__ scl = ws + WFRAG_FLOATS;
  const float* __restrict__ b1c = scl + 32;
  const float* __restrict__ cc  = b1c + 32;
  const float* __restrict__ bcv = cc + 32;

  const int tid  = threadIdx.x;
  const int wave = tid >> 5;
  const int lane = tid & 31;
  const int half = lane >> 4;     // 0: K even-pair / rows 0..7 of C; 1: odd / 8..15
  const int l16  = lane & 15;

  const int row0 = blockIdx.x * ROWS_PER_BLOCK + wave * 16;
  // A fragment: lane supplies x[row = l16][col = 4k + 2*half + j]
  const float* __restrict__ myrow = x + (size_t)(row0 + l16) * IN_DIM + 2 * half;
  const float* __restrict__ wlane = wfrag + lane * 2;

  v8f acc0 = {};
  v8f acc1 = {};

  #pragma unroll 4
  for (int k = 0; k < KSTEPS - 1; ++k) {
    v2f a;
    a.x = myrow[4 * k + 0];
    a.y = myrow[4 * k + 1];
    v2f b0, b1v;
    b0.x  = wlane[k * 128 + 0];
    b0.y  = wlane[k * 128 + 1];
    b1v.x = wlane[k * 128 + 64];
    b1v.y = wlane[k * 128 + 65];
    acc0 = __builtin_amdgcn_wmma_f32_16x16x4_f32(false, a, false, b0,  (short)0, acc0, false, false);
    acc1 = __builtin_amdgcn_wmma_f32_16x16x4_f32(false, a, false, b1v, (short)0, acc1, false, false);
  }
  { // tail k = 56 covers K = 224..227; only K = 224 (half 0, j 0) is real data
    const int k = KSTEPS - 1;
    v2f a;
    a.x = (half == 0) ? myrow[4 * k] : 0.0f;
    a.y = 0.0f;
    v2f b0, b1v;
    b0.x  = wlane[k * 128 + 0];
    b0.y  = wlane[k * 128 + 1];
    b1v.x = wlane[k * 128 + 64];
    b1v.y = wlane[k * 128 + 65];
    acc0 = __builtin_amdgcn_wmma_f32_16x16x4_f32(false, a, false, b0,  (short)0, acc0, false, false);
    acc1 = __builtin_amdgcn_wmma_f32_16x16x4_f32(false, a, false, b1v, (short)0, acc1, false, false);
  }

  // Epilogue: per-lane column n = 16t + l16 over 8 rows (C layout: VGPR v ->
  // row v for lanes 0..15, row v+8 for lanes 16..31).
  const float bconst = bcv[0];
  float p[8];
  {
    const int n0 = l16;
    const int n1 = 16 + l16;
    const float s0 = scl[n0], bb0 = b1c[n0], c0 = cc[n0];
    const float s1 = scl[n1], bb1 = b1c[n1], c1 = cc[n1];
    #pragma unroll
    for (int v = 0; v < 8; ++v) {
      const float h0 = acc0[v] * s0 + bb0;
      const float g0 = 0.5f * h0 * (1.0f + erff(h0 * 0.70710678118654752f));
      const float h1 = acc1[v] * s1 + bb1;
      const float g1 = 0.5f * h1 * (1.0f + erff(h1 * 0.70710678118654752f));
      p[v] = g0 * c0 + g1 * c1;
    }
  }

  // Sum the 28 weighted columns: reduce across each 16-lane half (masks 1..8
  // never cross halves under wave32).
  #pragma unroll
  for (int v = 0; v < 8; ++v) {
    float s = p[v];
    s += __shfl_xor(s, 1, 32);
    s += __shfl_xor(s, 2, 32);
    s += __shfl_xor(s, 4, 32);
    s += __shfl_xor(s, 8, 32);
    p[v] = s;
  }

  // Lanes 0..7 write rows row0+0..7, lanes 16..23 write rows row0+8..15.
  if (l16 < 8) {
    float r = p[0];
    #pragma unroll
    for (int v = 1; v < 8; ++v) r = (l16 == v) ? p[v] : r;
    r += bconst;
    const float sp = (r > 20.0f) ? r : log1pf(expf(r));
    out[row0 + half * 8 + l16] = sp;
  }
}

extern "C" void kernel_launch(void* const* d_in, const int* in_sizes, int n_in,
                              void* d_out, int out_size, void* d_ws, size_t ws_size,
                              hipStream_t stream)
{
  (void)in_sizes; (void)n_in; (void)ws_size;
  const float* x  = (const float*)d_in[0];
  const float* W1 = (const float*)d_in[1];
  const float* b1 = (const float*)d_in[2];
  const float* W2 = (const float*)d_in[3];
  const float* b2 = (const float*)d_in[4];
  float* out = (float*)d_out;
  float* ws  = (float*)d_ws;   // needs 7393 floats (~29 KB)

  prep_kernel<<<1, THREADS, 0, stream>>>(W1, b1, W2, b2, ws);
  const int nblocks = out_size / ROWS_PER_BLOCK;   // 262144 / 128 = 2048
  mlp_wmma_kernel<<<nblocks, THREADS, 0, stream>>>(x, ws, out);
}